// TransposeDiscriminator_38783554682983
// MI455X (gfx1250) — compile-verified
//
#include <hip/hip_runtime.h>

typedef float v2f __attribute__((ext_vector_type(2)));
typedef float v8f __attribute__((ext_vector_type(8)));

#define B_DIM 512
#define F_DIM 2048

// One workgroup per feature f. 256 threads = 8 wave32s.
// Each wave handles 4 blocks of 16 rows (8*4*16 = 512 rows).
// Per WMMA: one global_load_b64 per lane lands W directly in the f32 A-operand
// layout (lanes 0-15: K=0,1; lanes 16-31: K=2,3) -- no cross-lane shuffles.
// Outer K loop kept rolled (64 elems / 16 WMMAs per trip) so VGPR use stays
// < 128 -> 8 waves/SIMD of latency hiding instead of full-unroll's ~3.
__global__ __launch_bounds__(256)
__attribute__((amdgpu_waves_per_eu(8)))
void tiny_mlp_batched_matvec_kernel(
    const float* __restrict__ x,    // [B, F]
    const float* __restrict__ W1,   // [F, B, B]
    const float* __restrict__ b1,   // [F, B]
    const float* __restrict__ W2,   // [F, B]
    const float* __restrict__ b2,   // [F]
    float* __restrict__ out)        // [F]
{
    __shared__ float xs[B_DIM];       // x[:, f]
    __shared__ float partials[64];    // 8 waves * 4 blocks * 2 halves

    const int f    = blockIdx.x;
    const int tid  = threadIdx.x;
    const int lane = tid & 31;
    const int wave = tid >> 5;
    const int half = lane >> 4;       // 0: lanes 0-15, 1: lanes 16-31
    const int mrow = lane & 15;       // M index within 16-row tile

    // Stage x[:, f] into LDS (x is only 4MB total -> L2 resident)
    for (int j = tid; j < B_DIM; j += 256)
        xs[j] = x[(size_t)j * F_DIM + f];
    __syncthreads();

    const float* Wf = W1 + (size_t)f * B_DIM * B_DIM;

    for (int blk = 0; blk < 4; ++blk) {
        const int rowbase = wave * 64 + blk * 16;
        // Each lane's base: its row, offset +2 floats for the upper half so the
        // b64 load delivers exactly this lane's A-operand K pair.
        const float* lanePtr = Wf + (size_t)(rowbase + mrow) * B_DIM + half * 2;
        const float* rowPtr  = Wf + (size_t)(rowbase + mrow) * B_DIM;

        v8f c0 = {0.f, 0.f, 0.f, 0.f, 0.f, 0.f, 0.f, 0.f};
        v8f c1 = {0.f, 0.f, 0.f, 0.f, 0.f, 0.f, 0.f, 0.f};

        #pragma unroll 1
        for (int j0 = 0; j0 < B_DIM; j0 += 64) {
            // keep this row's stream ~1KB ahead
            __builtin_prefetch(rowPtr + j0 + 256, 0, 3);

            #pragma unroll
            for (int k = 0; k < 64; k += 8) {
                const int j = j0 + k;

                // A 16x4 f32 tiles: lane L<16 -> W[row][j..j+1] (K0,K1),
                //                   lane L>=16 -> W[row][j+2..j+3] (K2,K3).
                const float2 wA = *(const float2*)(lanePtr + j);
                const float2 wB = *(const float2*)(lanePtr + j + 4);
                v2f a0;  a0.x = wA.x;  a0.y = wA.y;
                v2f a1;  a1.x = wB.x;  a1.y = wB.y;

                // B 4x16 f32: K rows striped across lane halves per VGPR; every
                // N column carries the same x value so all D columns accumulate
                // the identical partial dot h[row].
                v2f bb0, bb1;
                bb0.x = xs[j + half];        bb0.y = xs[j + 2 + half];
                bb1.x = xs[j + 4 + half];    bb1.y = xs[j + 6 + half];

                // two independent accumulators -> no serialization on C chain
                c0 = __builtin_amdgcn_wmma_f32_16x16x4_f32(false, a0, false, bb0,
                                                           (short)0, c0, false, false);
                c1 = __builtin_amdgcn_wmma_f32_16x16x4_f32(false, a1, false, bb1,
                                                           (short)0, c1, false, false);
            }
        }

        const v8f c = c0 + c1;

        // C layout: VGPR r = row rowbase+r (lanes 0-15) / rowbase+8+r (lanes
        // 16-31), identical across columns. One lane per half finishes 8 rows.
        if (mrow == 0) {
            const int rbase = rowbase + half * 8;
            float partial = 0.f;
            #pragma unroll
            for (int r = 0; r < 8; ++r) {
                const int row = rbase + r;
                float h = c[r] + b1[(size_t)f * B_DIM + row];
                h = (h > 0.f) ? h : 0.2f * h;               // leaky_relu, slope 0.2
                partial += h * W2[(size_t)f * B_DIM + row];
            }
            partials[wave * 8 + blk * 2 + half] = partial;
        }
    }
    __syncthreads();

    // Deterministic fixed-order reduction
    if (tid == 0) {
        float s = b2[f];
        #pragma unroll 8
        for (int i = 0; i < 64; ++i) s += partials[i];
        out[f] = s;
    }
}

extern "C" void kernel_launch(void* const* d_in, const int* in_sizes, int n_in,
                              void* d_out, int out_size, void* d_ws, size_t ws_size,
                              hipStream_t stream) {
    const float* x  = (const float*)d_in[0];
    const float* W1 = (const float*)d_in[1];
    const float* b1 = (const float*)d_in[2];
    const float* W2 = (const float*)d_in[3];
    const float* b2 = (const float*)d_in[4];
    float* out = (float*)d_out;

    dim3 grid(F_DIM);
    dim3 block(256);
    tiny_mlp_batched_matvec_kernel<<<grid, block, 0, stream>>>(x, W1, b1, W2, b2, out);
}